// FocalLoss_79955111182917
// MI455X (gfx1250) — compile-verified
//
#include <hip/hip_runtime.h>
#include <cstdint>
#include <cstddef>

// ---------------------------------------------------------------------------
// FocalLoss for MI455X (gfx1250, wave32).
//
// d^2(anchor, gt) = |a|^2 + |g|^2 - 2 a.g  is a K=4 bilinear form:
//   A row (16x4)  = [ax, ay, ax^2+ay^2, 1]
//   B col (4x16)  = [-2gx, -2gy, 1, gx^2+gy^2]        (invalid gt -> [0,0,1,1e7])
// so V_WMMA_F32_16X16X4_F32 produces a 16x16 tile of squared distances.
// C accumulator = 1.0 biases d^2 by +1 so cancellation can never make it
// negative, keeping the "float bits are uint-ordered" argmin-key trick valid:
//   key = (bits(d2+1) & ~63) | annotation_index   -> argmin == v_min_u32.
// ---------------------------------------------------------------------------

typedef float v2f __attribute__((ext_vector_type(2)));
typedef float v8f __attribute__((ext_vector_type(8)));

#define BB 4
#define NN 262144
#define MM 64
#define BLOCK_ANCH 2048
#define BLOCKS_PER_IMG (NN / BLOCK_ANCH)   // 128
#define THREADS 256
#define WAVES (THREADS / 32)               // 8 wave32s
#define ANCH_PER_WAVE (BLOCK_ANCH / WAVES) // 256
#define ITERS (ANCH_PER_WAVE / 32)         // 8

__device__ __forceinline__ unsigned umin32(unsigned a, unsigned b) { return a < b ? a : b; }

// wave32 sum via ds_swizzle XOR patterns (group-of-32 mode: and=0x1f, xor=mask<<10)
__device__ __forceinline__ float wave_sum32(float v) {
    v += __int_as_float(__builtin_amdgcn_ds_swizzle(__float_as_int(v), (1  << 10) | 31));
    v += __int_as_float(__builtin_amdgcn_ds_swizzle(__float_as_int(v), (2  << 10) | 31));
    v += __int_as_float(__builtin_amdgcn_ds_swizzle(__float_as_int(v), (4  << 10) | 31));
    v += __int_as_float(__builtin_amdgcn_ds_swizzle(__float_as_int(v), (8  << 10) | 31));
    v += __int_as_float(__builtin_amdgcn_ds_swizzle(__float_as_int(v), (16 << 10) | 31));
    return v;
}

__global__ void ws_init_kernel(float* ws) {
    if (threadIdx.x < 4 * BB) ws[threadIdx.x] = 0.0f;
}

__global__ __launch_bounds__(THREADS)
void focal_main_kernel(const float* __restrict__ cls,   // (B,N,4)
                       const float* __restrict__ reg,   // (B,N,3)
                       const float* __restrict__ anc,   // (B,N,3) -> use image 0
                       const float* __restrict__ ann,   // (B,64,5)
                       float* __restrict__ ws)          // (B,4): npos, cls, xy, ang
{
    __shared__ float s_gx[MM], s_gy[MM], s_gal[MM], s_lbl[MM];
    __shared__ float s_m2gx[MM], s_m2gy[MM], s_g2s[MM];
    __shared__ unsigned s_key[WAVES][32][17];           // stride 17 -> no bank conflicts

    const int tid = threadIdx.x;
    const int b   = blockIdx.x / BLOCKS_PER_IMG;
    const int blk = blockIdx.x % BLOCKS_PER_IMG;

    // Stage the 64 annotations + derived WMMA B-columns in LDS.
    if (tid < MM) {
        const float* a5 = ann + (b * MM + tid) * 5;
        float gx = a5[0], gy = a5[1], gal = a5[2], lb = a5[3];
        bool valid = (lb != -1.0f);
        s_gx[tid] = gx; s_gy[tid] = gy; s_gal[tid] = gal; s_lbl[tid] = lb;
        s_m2gx[tid] = valid ? -2.0f * gx : 0.0f;
        s_m2gy[tid] = valid ? -2.0f * gy : 0.0f;
        s_g2s[tid]  = valid ? gx * gx + gy * gy : 1.0e7f;  // sentinel >> max real d^2 (2e6)
    }
    __syncthreads();

    const int  wave = tid >> 5;
    const int  lane = tid & 31;
    const int  ln   = lane & 15;
    const bool hi   = lane >= 16;

    // B tiles (4x16 f32, K split low/high lane halves like the A-matrix table):
    //   vgpr0: lanes0-15 K=0 (-2gx), lanes16-31 K=2 (1.0)
    //   vgpr1: lanes0-15 K=1 (-2gy), lanes16-31 K=3 (|g|^2 or sentinel)
    v2f Bt[4];
#pragma unroll
    for (int t = 0; t < 4; ++t) {
        int j = t * 16 + ln;
        Bt[t][0] = hi ? 1.0f     : s_m2gx[j];
        Bt[t][1] = hi ? s_g2s[j] : s_m2gy[j];
    }
    v8f cOne;
#pragma unroll
    for (int v = 0; v < 8; ++v) cOne[v] = 1.0f;          // +1 bias keeps d^2+1 > 0

    float acc_np = 0.f, acc_cls = 0.f, acc_xy = 0.f, acc_ang = 0.f;

    const int waveBase = blk * BLOCK_ANCH + wave * ANCH_PER_WAVE;
    unsigned* kb = &s_key[wave][0][0];

    for (int it = 0; it < ITERS; ++it) {
        const int abase32 = waveBase + it * 32;

        // prefetch next iteration's classification cachelines into GL2/L0
        __builtin_prefetch(cls + ((size_t)b * NN + (abase32 + 32)) * 4, 0, 1);

        // Two 16-anchor WMMA groups -> 32 anchors per wave iteration.
#pragma unroll
        for (int g = 0; g < 2; ++g) {
            const int arow = abase32 + g * 16 + ln;       // A-matrix row anchor
            float ax = anc[arow * 3 + 0];
            float ay = anc[arow * 3 + 1];
            v2f A;
            A[0] = hi ? (ax * ax + ay * ay) : ax;         // K=0 / K=2
            A[1] = hi ? 1.0f : ay;                        // K=1 / K=3

            unsigned km[8];
#pragma unroll
            for (int v = 0; v < 8; ++v) km[v] = 0xFFFFFFFFu;

#pragma unroll
            for (int t = 0; t < 4; ++t) {
                // D = A*B + 1 : 16x16 tile of (d^2 + 1)
                v8f D = __builtin_amdgcn_wmma_f32_16x16x4_f32(
                            false, A, false, Bt[t], (short)0, cOne, false, false);
#pragma unroll
                for (int v = 0; v < 8; ++v) {
                    unsigned k = (__float_as_uint(D[v]) & 0xFFFFFFC0u) | (unsigned)(t << 4);
                    km[v] = umin32(km[v], k);             // argmin over tiles via u32 min
                }
            }
            // D layout: vgpr v, lanes0-15 -> row v, lanes16-31 -> row v+8; lane = column.
            const int rbase = g * 16 + (hi ? 8 : 0);
#pragma unroll
            for (int v = 0; v < 8; ++v)
                kb[(rbase + v) * 17 + ln] = km[v] | (unsigned)ln;  // low 6 bits = gt index
        }

        // intra-wave LDS store -> load handoff (LDS is in-order per wave)
        asm volatile("s_wait_dscnt 0" ::: "memory");

        // Each lane now owns anchor (abase32 + lane): min its row of 16 partial keys.
        {
            const unsigned* rowp = kb + lane * 17;
            unsigned key = rowp[0];
#pragma unroll
            for (int j = 1; j < 16; ++j) key = umin32(key, rowp[j]);

            const int   a   = abase32 + lane;
            const int   idx = (int)(key & 63u);
            const float d2b = __uint_as_float(key & 0xFFFFFFC0u);  // min d^2 + 1

            float ax = anc[a * 3 + 0], ay = anc[a * 3 + 1], aal = anc[a * 3 + 2];
            float gx = s_gx[idx], gy = s_gy[idx], gal = s_gal[idx], lb = s_lbl[idx];
            float aang = fabsf(aal - gal);

            // dxy<=10 <-> d2+1<=101 ; dxy>=15 <-> d2+1>=226
            bool positive = (d2b <= 101.0f) && (aang <= 15.0f) && (lb != -1.0f);
            bool faraway  = (d2b >= 226.0f) || (aang >= 22.5f);

            // ---- focal classification loss over C=4 ----
            const float4 c4 = *(const float4*)(cls + ((size_t)b * NN + a) * 4);
            float cv[4] = {c4.x, c4.y, c4.z, c4.w};
            int li = (int)fminf(fmaxf(lb, 0.0f), 3.0f);
            float clsum = 0.0f;
#pragma unroll
            for (int i = 0; i < 4; ++i) {
                float c = fminf(fmaxf(cv[i], 1.0e-4f), 1.0f - 1.0e-4f);
                bool t1 = positive && (i == li);
                float omc = 1.0f - c;
                float pos_t = 0.95f * omc * omc * (-__logf(c));    // t==1
                float neg_t = 0.05f * c * c * (-__logf(omc));      // t==0
                clsum += t1 ? pos_t : neg_t;
            }
            clsum = (positive || faraway) ? clsum : 0.0f;          // t==-1 -> ignored

            // ---- regression losses ----
            const float* r3 = reg + ((size_t)b * NN + a) * 3;
            float rx = r3[0], ry = r3[1], ra = r3[2];
            float tx = gx - ax, ty = gy - ay, ta = gal - aal;
            float dxr = fabsf(tx - rx), dyr = fabsf(ty - ry);
            float sx = (dxr <= (1.0f / 9.0f)) ? 4.5f * dxr * dxr : dxr - (1.0f / 18.0f);
            float sy = (dyr <= (1.0f / 9.0f)) ? 4.5f * dyr * dyr : dyr - (1.0f / 18.0f);
            float da = fmaxf((fabsf(ta - ra) - 10.0f) * 0.2f, 0.0f);

            if (positive) { acc_np += 1.0f; acc_xy += sx + sy; acc_ang += da; }
            acc_cls += clsum;
        }
    }

    acc_np  = wave_sum32(acc_np);
    acc_cls = wave_sum32(acc_cls);
    acc_xy  = wave_sum32(acc_xy);
    acc_ang = wave_sum32(acc_ang);
    if (lane == 0) {
        atomicAdd(&ws[b * 4 + 0], acc_np);
        atomicAdd(&ws[b * 4 + 1], acc_cls);
        atomicAdd(&ws[b * 4 + 2], acc_xy);
        atomicAdd(&ws[b * 4 + 3], acc_ang);
    }
}

__global__ void finalize_kernel(const float* __restrict__ ws, float* __restrict__ out) {
    if (threadIdx.x == 0) {
        float c = 0.f, x = 0.f, g = 0.f;
        for (int b = 0; b < BB; ++b) {
            float np = fmaxf(ws[b * 4 + 0], 1.0f);
            c += ws[b * 4 + 1] / np;
            x += ws[b * 4 + 2] / (2.0f * np);   // xy sums are 0 when npos==0 -> has_pos ok
            g += ws[b * 4 + 3] / np;
        }
        out[0] = c * 0.25f;
        out[1] = x * 0.25f;
        out[2] = g * 0.25f;
    }
}

extern "C" void kernel_launch(void* const* d_in, const int* in_sizes, int n_in,
                              void* d_out, int out_size, void* d_ws, size_t ws_size,
                              hipStream_t stream) {
    const float* cls = (const float*)d_in[0];
    const float* reg = (const float*)d_in[1];
    const float* anc = (const float*)d_in[2];   // (B,N,3), all images identical -> use [0]
    const float* ann = (const float*)d_in[3];
    float* ws  = (float*)d_ws;
    float* out = (float*)d_out;

    ws_init_kernel<<<1, 32, 0, stream>>>(ws);
    focal_main_kernel<<<BB * BLOCKS_PER_IMG, THREADS, 0, stream>>>(cls, reg, anc, ann, ws);
    finalize_kernel<<<1, 32, 0, stream>>>(ws, out);
}